// MSGNN_link_prediction_54778012893400
// MI455X (gfx1250) — compile-verified
//
#include <hip/hip_runtime.h>
#include <math.h>

// ---- problem constants (from reference) ----
#define NN     2000
#define NE     64000
#define NQ     200000
#define F_IN   512
#define HH     64
#define SIMD_  64
#define PI_F   3.14159265358979323846f

typedef float v2f __attribute__((ext_vector_type(2)));
typedef float v8f __attribute__((ext_vector_type(8)));

// ---------------------------------------------------------------------------
// 1) degree accumulation: deg[dst]+=0.5|w|, deg[src]+=0.5|w|
// ---------------------------------------------------------------------------
__global__ __launch_bounds__(256)
void deg_kernel(const int* __restrict__ ei, const float* __restrict__ w,
                float* __restrict__ deg, int E) {
  int e = blockIdx.x * blockDim.x + threadIdx.x;
  if (e >= E) return;
  float hw = 0.5f * fabsf(w[e]);
  atomicAdd(&deg[ei[E + e]], hw);   // dst
  atomicAdd(&deg[ei[e]],     hw);   // src
}

// ---------------------------------------------------------------------------
// 2) dinv = deg>0 ? 1/sqrt(deg) : 0
// ---------------------------------------------------------------------------
__global__ __launch_bounds__(256)
void dinv_kernel(const float* __restrict__ deg, float* __restrict__ dinv, int n) {
  int i = blockIdx.x * blockDim.x + threadIdx.x;
  if (i >= n) return;
  float d = deg[i];
  dinv[i] = (d > 0.0f) ? (1.0f / sqrtf(d)) : 0.0f;
}

// ---------------------------------------------------------------------------
// 3) per-directed-edge complex coefficients (layer-invariant, computed once)
//    forward  (de=2e):   s=src,d=dst, theta=+ (pi/2) w
//    backward (de=2e+1): s=dst,d=src, theta=- (pi/2) w
//    cr = -norm*cos(theta), ci = -norm*sin(theta), norm=0.5 w dinv[s] dinv[d]
// ---------------------------------------------------------------------------
__global__ __launch_bounds__(256)
void coef_kernel(const int* __restrict__ ei, const float* __restrict__ w,
                 const float* __restrict__ dinv,
                 float* __restrict__ cr, float* __restrict__ ci, int E) {
  int e = blockIdx.x * blockDim.x + threadIdx.x;
  if (e >= E) return;
  int s = ei[e], d = ei[E + e];
  float we   = w[e];
  float norm = 0.5f * we * dinv[s] * dinv[d];
  float th   = (0.5f * PI_F) * we;          // 2*pi*0.25*w
  float c = cosf(th), sn = sinf(th);
  cr[2 * e]     = -norm * c;
  ci[2 * e]     = -norm * sn;
  cr[2 * e + 1] = -norm * c;
  ci[2 * e + 1] =  norm * sn;               // sin(-th) negated
}

// ---------------------------------------------------------------------------
// 4) complex message scatter:  t1[d] += (cr + i ci) * x[s]   (per feature)
//    one thread = one directed edge x 4 consecutive features
// ---------------------------------------------------------------------------
__global__ __launch_bounds__(256)
void scatter_kernel(const int* __restrict__ ei,
                    const float* __restrict__ cr, const float* __restrict__ ci,
                    const float* __restrict__ xr, const float* __restrict__ xi,
                    float* __restrict__ tr, float* __restrict__ ti,
                    int E, int F, int chunks, long total) {
  long gtid = (long)blockIdx.x * blockDim.x + threadIdx.x;
  if (gtid >= total) return;
  int c  = (int)(gtid % chunks);
  int de = (int)(gtid / chunks);
  int e  = de >> 1;
  bool fwd = ((de & 1) == 0);
  int s = fwd ? ei[e]     : ei[E + e];
  int d = fwd ? ei[E + e] : ei[e];
  float c_r = cr[de], c_i = ci[de];
  int f = c * 4;
  const float4 a = *(const float4*)(xr + (size_t)s * F + f);
  const float4 b = *(const float4*)(xi + (size_t)s * F + f);
  float* trd = tr + (size_t)d * F + f;
  float* tid = ti + (size_t)d * F + f;
  atomicAdd(trd + 0, c_r * a.x - c_i * b.x);
  atomicAdd(tid + 0, c_r * b.x + c_i * a.x);
  atomicAdd(trd + 1, c_r * a.y - c_i * b.y);
  atomicAdd(tid + 1, c_r * b.y + c_i * a.y);
  atomicAdd(trd + 2, c_r * a.z - c_i * b.z);
  atomicAdd(tid + 2, c_r * b.z + c_i * a.z);
  atomicAdd(trd + 3, c_r * a.w - c_i * b.w);
  atomicAdd(tid + 3, c_r * b.w + c_i * a.w);
}

// ---------------------------------------------------------------------------
// 5) fused dense transform + ReLU mask, via V_WMMA_F32_16X16X4_F32 (wave32).
//    One wave computes a full 16-row x H(=64)-col output strip:
//      4 N-tiles x (real, imag) = 8 v8f accumulators, A fragments loaded once
//      per k-step and reused across all 4 N-tiles (16 WMMAs / k-step).
//    yr = relu_mask .* (xr@W0 + tr@W1),  yi = relu_mask .* (xi@W0 + ti@W1)
//    Fragment layouts per CDNA5 ISA 7.12.2 (f32 16x16x4):
//      A:   lane l -> m=l&15, koff=(l>>4)*2 : {A[m][koff], A[m][koff+1]}
//      B:   lane l -> n=l&15, koff=(l>>4)*2 : {B[koff][n], B[koff+1][n]}
//      C/D: vgpr r -> D[(l>>4)*8 + r][l&15]
// ---------------------------------------------------------------------------
__global__ __launch_bounds__(32)
void gemm_relu_kernel(const float* __restrict__ xr, const float* __restrict__ xi,
                      const float* __restrict__ tr, const float* __restrict__ ti,
                      const float* __restrict__ W,   // (2, F, H)
                      float* __restrict__ yr, float* __restrict__ yi,
                      int F, int H) {
  const int lane = threadIdx.x;
  const int mrow = lane & 15;
  const int half = lane >> 4;                 // 0 or 1
  const int m    = blockIdx.x * 16 + mrow;    // A row (same for both halves)
  const float* W0 = W;
  const float* W1 = W + (size_t)F * H;

  v8f accR[4], accI[4];
#pragma unroll
  for (int nt = 0; nt < 4; ++nt) {
    accR[nt] = (v8f){0.f, 0.f, 0.f, 0.f, 0.f, 0.f, 0.f, 0.f};
    accI[nt] = (v8f){0.f, 0.f, 0.f, 0.f, 0.f, 0.f, 0.f, 0.f};
  }

  for (int k = 0; k < F; k += 4) {
    const int ka = k + half * 2;
    v2f a_xr = *(const v2f*)(xr + (size_t)m * F + ka);
    v2f a_tr = *(const v2f*)(tr + (size_t)m * F + ka);
    v2f a_xi = *(const v2f*)(xi + (size_t)m * F + ka);
    v2f a_ti = *(const v2f*)(ti + (size_t)m * F + ka);
#pragma unroll
    for (int nt = 0; nt < 4; ++nt) {
      const int n = nt * 16 + mrow;           // B col for this N-tile
      v2f b0, b1;
      b0[0] = W0[(size_t)ka * H + n];
      b0[1] = W0[(size_t)(ka + 1) * H + n];
      b1[0] = W1[(size_t)ka * H + n];
      b1[1] = W1[(size_t)(ka + 1) * H + n];
      accR[nt] = __builtin_amdgcn_wmma_f32_16x16x4_f32(false, a_xr, false, b0,
                                                       (short)0, accR[nt],
                                                       false, false);
      accR[nt] = __builtin_amdgcn_wmma_f32_16x16x4_f32(false, a_tr, false, b1,
                                                       (short)0, accR[nt],
                                                       false, false);
      accI[nt] = __builtin_amdgcn_wmma_f32_16x16x4_f32(false, a_xi, false, b0,
                                                       (short)0, accI[nt],
                                                       false, false);
      accI[nt] = __builtin_amdgcn_wmma_f32_16x16x4_f32(false, a_ti, false, b1,
                                                       (short)0, accI[nt],
                                                       false, false);
    }
  }

  // store with ReLU mask derived from real part (applied to both parts)
#pragma unroll
  for (int nt = 0; nt < 4; ++nt) {
#pragma unroll
    for (int r = 0; r < 8; ++r) {
      int mm = blockIdx.x * 16 + half * 8 + r;
      int nn = nt * 16 + mrow;
      float vr = accR[nt][r];
      float vi = accI[nt][r];
      float keep = (vr >= 0.0f) ? 1.0f : 0.0f;
      yr[(size_t)mm * H + nn] = vr * keep;
      yi[(size_t)mm * H + nn] = vi * keep;
    }
  }
}

// ---------------------------------------------------------------------------
// 6) query stage: one wave32 per query.
//    x = [hr[ni](64) | hr[nj](64) | hi[ni](64) | hi[nj](64) | sim(64)]  (320)
//    logits = x @ lin_w + lin_b ; out = log_softmax(logits)
//    sim rows are a 51 MB random gather from a 1 GB table (the HBM-dominant
//    term) -> kick off a global_prefetch as early as possible.
// ---------------------------------------------------------------------------
__global__ __launch_bounds__(256)
void query_kernel(const int* __restrict__ qe,
                  const float* __restrict__ hr, const float* __restrict__ hi,
                  const float* __restrict__ emb,
                  const float* __restrict__ lw, const float* __restrict__ lb,
                  float* __restrict__ out, int Q) {
  int wave = (blockIdx.x * blockDim.x + threadIdx.x) >> 5;
  int lane = threadIdx.x & 31;
  if (wave >= Q) return;
  int ni = qe[2 * wave];
  int nj = qe[2 * wave + 1];
  const float* sim = emb + ((size_t)ni * NN + nj) * SIMD_;
  // 256B row: prefetch both cachelines (lanes 0 and 16 cover +0 / +128B)
  __builtin_prefetch(sim + (lane & 16) * 2, 0, 1);   // global_prefetch_b8

  float a0 = 0.0f, a1 = 0.0f;
#pragma unroll
  for (int j = 0; j < 10; ++j) {        // 320 features / 32 lanes
    int f = lane + 32 * j;              // segment boundaries are wave-uniform
    float x;
    if      (f < 64)  x = hr[(size_t)ni * HH + f];
    else if (f < 128) x = hr[(size_t)nj * HH + (f - 64)];
    else if (f < 192) x = hi[(size_t)ni * HH + (f - 128)];
    else if (f < 256) x = hi[(size_t)nj * HH + (f - 192)];
    else              x = sim[f - 256];
    a0 += x * lw[2 * f + 0];
    a1 += x * lw[2 * f + 1];
  }
  // wave32 butterfly reduction
#pragma unroll
  for (int off = 16; off > 0; off >>= 1) {
    a0 += __shfl_xor(a0, off, 32);
    a1 += __shfl_xor(a1, off, 32);
  }
  if (lane == 0) {
    float l0 = a0 + lb[0];
    float l1 = a1 + lb[1];
    float mx  = fmaxf(l0, l1);
    float lse = mx + logf(expf(l0 - mx) + expf(l1 - mx));
    out[2 * wave]     = l0 - lse;
    out[2 * wave + 1] = l1 - lse;
  }
}

// ---------------------------------------------------------------------------
extern "C" void kernel_launch(void* const* d_in, const int* in_sizes, int n_in,
                              void* d_out, int out_size, void* d_ws, size_t ws_size,
                              hipStream_t stream) {
  const float* real = (const float*)d_in[0];     // (N, F_IN)
  const float* imag = (const float*)d_in[1];     // (N, F_IN)
  const int*   ei   = (const int*)  d_in[2];     // (2, E): src=[0,E), dst=[E,2E)
  const int*   qe   = (const int*)  d_in[3];     // (Q, 2)
  const float* emb  = (const float*)d_in[4];     // (N*N, SIM)
  const float* w    = (const float*)d_in[5];     // (E,)
  const float* W1   = (const float*)d_in[6];     // (2, F_IN, H)
  const float* W2   = (const float*)d_in[7];     // (2, H, H)
  const float* lw   = (const float*)d_in[8];     // (4H+SIM, 2)
  const float* lb   = (const float*)d_in[9];     // (2,)
  float* out = (float*)d_out;

  // ---- workspace layout (floats), all 16B aligned chunk sizes ----
  float* ws   = (float*)d_ws;
  float* deg  = ws;                      size_t o = 2048;
  float* dinv = ws + o;                  o += 2048;
  float* cr   = ws + o;                  o += 2 * NE;
  float* ci   = ws + o;                  o += 2 * NE;
  float* t1r1 = ws + o;                  o += (size_t)NN * F_IN;
  float* t1i1 = ws + o;                  o += (size_t)NN * F_IN;
  float* h1r  = ws + o;                  o += (size_t)NN * HH;
  float* h1i  = ws + o;                  o += (size_t)NN * HH;
  float* t1r2 = ws + o;                  o += (size_t)NN * HH;
  float* t1i2 = ws + o;                  o += (size_t)NN * HH;
  float* h2r  = ws + o;                  o += (size_t)NN * HH;
  float* h2i  = ws + o;                  o += (size_t)NN * HH;
  (void)ws_size; (void)in_sizes; (void)n_in; (void)out_size;

  // zero atomic accumulators (capture-safe async memsets)
  hipMemsetAsync(deg,  0, 2048 * sizeof(float), stream);
  hipMemsetAsync(t1r1, 0, 2 * (size_t)NN * F_IN * sizeof(float), stream); // t1r1+t1i1
  hipMemsetAsync(t1r2, 0, 2 * (size_t)NN * HH * sizeof(float), stream);   // t1r2+t1i2

  // graph normalization + phase coefficients (shared by both layers)
  deg_kernel <<<(NE + 255) / 256, 256, 0, stream>>>(ei, w, deg, NE);
  dinv_kernel<<<(NN + 255) / 256, 256, 0, stream>>>(deg, dinv, NN);
  coef_kernel<<<(NE + 255) / 256, 256, 0, stream>>>(ei, w, dinv, cr, ci, NE);

  // ---- layer 1 ----
  {
    long total = 2L * NE * (F_IN / 4);
    scatter_kernel<<<(int)((total + 255) / 256), 256, 0, stream>>>(
        ei, cr, ci, real, imag, t1r1, t1i1, NE, F_IN, F_IN / 4, total);
    gemm_relu_kernel<<<NN / 16, 32, 0, stream>>>(real, imag, t1r1, t1i1, W1,
                                                 h1r, h1i, F_IN, HH);
  }

  // ---- layer 2 ----
  {
    long total = 2L * NE * (HH / 4);
    scatter_kernel<<<(int)((total + 255) / 256), 256, 0, stream>>>(
        ei, cr, ci, h1r, h1i, t1r2, t1i2, NE, HH, HH / 4, total);
    gemm_relu_kernel<<<NN / 16, 32, 0, stream>>>(h1r, h1i, t1r2, t1i2, W2,
                                                 h2r, h2i, HH, HH);
  }

  // ---- query stage: wave32 per query ----
  {
    long threads = 32L * NQ;
    query_kernel<<<(int)((threads + 255) / 256), 256, 0, stream>>>(
        qe, h2r, h2i, emb, lw, lb, out, NQ);
  }
}